// MultiLayerGCN_37417755083137
// MI455X (gfx1250) — compile-verified
//
#include <hip/hip_runtime.h>

// ---------------------------------------------------------------------------
// MultiLayerGCN for MI455X (gfx1250, wave32)
//   GEMMs:       native fp32 WMMA V_WMMA_F32_16X16X4_F32, B panel packed to
//                fragment layout and staged in LDS (ds_load_b64 fragments)
//   Aggregation: L2-resident global_atomic_add_f32 scatter (fits 192MB L2)
// ---------------------------------------------------------------------------

#define NODES   50000
#define EDGES   800000
#define FIN     128
#define HID     256
#define OUTC    40

typedef float v2f __attribute__((ext_vector_type(2)));
typedef float v8f __attribute__((ext_vector_type(8)));

// ---------------- utility kernels ----------------

__global__ void fill_f32(float* __restrict__ p, float v, long long n) {
    long long i = (long long)blockIdx.x * blockDim.x + threadIdx.x;
    if (i < n) p[i] = v;
}

__global__ void degree_count(const int* __restrict__ dst, float* __restrict__ deg, int e) {
    int i = blockIdx.x * blockDim.x + threadIdx.x;
    if (i < e) atomicAdd(&deg[dst[i]], 1.0f);
}

__global__ void rsqrt_inplace(float* __restrict__ d, int n) {
    int i = blockIdx.x * blockDim.x + threadIdx.x;
    if (i < n) d[i] = rsqrtf(fmaxf(d[i], 1.0f));
}

// ---------------- weight repack: B[K,N] -> Bp[(k4*Npad + n)*4 + j] ----------
// Groups each column's 4 consecutive K values contiguously so a lane's WMMA
// B-fragment is a single 8-byte load. Pads columns [N, Npad) with zeros.
__global__ void pack_B(const float* __restrict__ B, float* __restrict__ Bp,
                       int K, int N, int Npad) {
    int idx = blockIdx.x * blockDim.x + threadIdx.x;   // over K4*Npad
    int K4 = K >> 2;
    if (idx >= K4 * Npad) return;
    int k4 = idx / Npad;
    int n  = idx - k4 * Npad;
    float4 v = make_float4(0.f, 0.f, 0.f, 0.f);
    if (n < N) {
        const float* c = B + (size_t)(4 * k4) * N + n;
        v.x = c[0];
        v.y = c[(size_t)N];
        v.z = c[(size_t)2 * N];
        v.w = c[(size_t)3 * N];
    }
    ((float4*)Bp)[idx] = v;
}

// ---------------- fp32 WMMA GEMM: C[M,N] = A[M,K] * B[K,N] ----------------
// block = 256 threads = 8 waves; wave grid 4(M) x 2(N); wave tile 16 x 64.
// Block tile: 64 (M) x 128 (N). B is pre-packed (pack_B) and staged in LDS.
// Per 16x16x4 f32 WMMA (wave32):
//   A (16x4): lanes 0-15 -> {K0,K1}, lanes 16-31 -> {K2,K3}  (v2f per lane)
//   B (4x16): mirrored pair layout per lane-half                (v2f per lane)
//   C (16x16): vgpr v -> row v (lanes 0-15) / row v+8 (lanes 16-31), col=lane%16
__global__ void __launch_bounds__(256)
wmma_gemm_f32(const float* __restrict__ A, const float* __restrict__ Bp,
              float* __restrict__ C, int M, int N, int Npad, int K) {
    extern __shared__ float lds[];            // K4 * 128 * 4 floats
    const int K4   = K >> 2;
    const int lane = threadIdx.x & 31;
    const int wave = threadIdx.x >> 5;        // 0..7
    const int half = lane >> 4;               // 0 or 1
    const int r    = lane & 15;

    const int wm  = wave & 3;                 // wave position in M
    const int wn  = wave >> 2;                // wave position in N
    const int m0  = blockIdx.x * 64 + wm * 16;
    const int nb0 = blockIdx.y * 128;

    // stage this block's 128-column packed B panel into LDS (coalesced 16B)
    {
        const float4* gsrc = (const float4*)Bp;
        float4*       ldst = (float4*)lds;
        for (int i = threadIdx.x; i < K4 * 128; i += 256) {
            int k4 = i >> 7;
            int c  = i & 127;
            ldst[i] = gsrc[(size_t)k4 * Npad + nb0 + c];
        }
    }
    __syncthreads();

    v8f acc0 = {}, acc1 = {}, acc2 = {}, acc3 = {};

    // clamp row so every lane issues a valid load (WMMA requires full EXEC)
    const int arow = (m0 + r) < (M - 1) ? (m0 + r) : (M - 1);
    const float* Aptr = A + (size_t)arow * K + 2 * half;

    // lane-local LDS fragment bases (byte-offset friendly: +64 floats apart)
    const float* lb = lds + ((wn * 64 + r) << 2) + 2 * half;

    for (int k4 = 0; k4 < K4; ++k4) {
        v2f a = *(const v2f*)(Aptr + (k4 << 2));
        const float* l = lb + (k4 << 9);      // k4 * 128 cols * 4 floats
        v2f b0 = *(const v2f*)(l + (0  << 6));
        v2f b1 = *(const v2f*)(l + (1  << 6));
        v2f b2 = *(const v2f*)(l + (2  << 6));
        v2f b3 = *(const v2f*)(l + (3  << 6));

        acc0 = __builtin_amdgcn_wmma_f32_16x16x4_f32(false, a, false, b0, (short)0, acc0, false, false);
        acc1 = __builtin_amdgcn_wmma_f32_16x16x4_f32(false, a, false, b1, (short)0, acc1, false, false);
        acc2 = __builtin_amdgcn_wmma_f32_16x16x4_f32(false, a, false, b2, (short)0, acc2, false, false);
        acc3 = __builtin_amdgcn_wmma_f32_16x16x4_f32(false, a, false, b3, (short)0, acc3, false, false);
    }

    // guarded stores (padded B columns are zero; never stored beyond N)
    v8f* accs[4] = {&acc0, &acc1, &acc2, &acc3};
    #pragma unroll
    for (int t = 0; t < 4; ++t) {
        int col = nb0 + wn * 64 + t * 16 + r;
        if (col < N) {
            #pragma unroll
            for (int v = 0; v < 8; ++v) {
                int row = m0 + v + 8 * half;
                if (row < M) C[(size_t)row * N + col] = (*accs[t])[v];
            }
        }
    }
}

// ---------------- sparse aggregation: AGG[dst] += H[src] * dinv[s]*dinv[d] ----
// one thread per (edge, 4-channel group); 64 consecutive threads share an edge
// -> coalesced 1KB row gather, scalar f32 atomics land in L2 (AGG fits L2).
__global__ void scatter_edges(const float* __restrict__ H, float* __restrict__ AGG,
                              const int* __restrict__ src, const int* __restrict__ dst,
                              const float* __restrict__ dinv, int e, int F) {
    const int groups = F >> 2;
    long long idx = (long long)blockIdx.x * blockDim.x + threadIdx.x;
    long long total = (long long)e * groups;
    if (idx >= total) return;
    int ed = (int)(idx / groups);
    int g  = (int)(idx - (long long)ed * groups);
    int s = src[ed], d = dst[ed];
    float w = dinv[s] * dinv[d];
    const float4 h = *(const float4*)(H + (size_t)s * F + (g << 2));
    float* out = AGG + (size_t)d * F + (g << 2);
    atomicAdd(out + 0, h.x * w);
    atomicAdd(out + 1, h.y * w);
    atomicAdd(out + 2, h.z * w);
    atomicAdd(out + 3, h.w * w);
}

// ---------------- epilogue: AGG = relu(AGG + H*dinv^2 + bias) (self-loop fused)
__global__ void epilogue_relu(float* __restrict__ AGG, const float* __restrict__ H,
                              const float* __restrict__ dinv, const float* __restrict__ bias,
                              int n, int F) {
    long long idx = (long long)blockIdx.x * blockDim.x + threadIdx.x;
    if (idx >= (long long)n * F) return;
    int node = (int)(idx / F);
    int c    = (int)(idx - (long long)node * F);
    float di = dinv[node];
    float v = AGG[idx] + H[idx] * (di * di) + bias[c];
    AGG[idx] = fmaxf(v, 0.0f);
}

__global__ void add_bias(float* __restrict__ out, const float* __restrict__ b, int total, int F) {
    int i = blockIdx.x * blockDim.x + threadIdx.x;
    if (i < total) out[i] += b[i % F];
}

// ---------------- host-side orchestration ----------------

extern "C" void kernel_launch(void* const* d_in, const int* in_sizes, int n_in,
                              void* d_out, int out_size, void* d_ws, size_t ws_size,
                              hipStream_t stream) {
    const float* x  = (const float*)d_in[0];   // [NODES, FIN]
    const float* W1 = (const float*)d_in[1];   // [FIN, HID]
    const float* b1 = (const float*)d_in[2];   // [HID]
    const float* W2 = (const float*)d_in[3];   // [HID, HID]
    const float* b2 = (const float*)d_in[4];   // [HID]
    const float* Wc = (const float*)d_in[5];   // [HID, OUTC]
    const float* bc = (const float*)d_in[6];   // [OUTC]
    const int*   ei = (const int*)d_in[7];     // [2, EDGES]
    const int* srcp = ei;
    const int* dstp = ei + EDGES;

    // workspace: two ping-pong node-feature buffers + dinv + packed weights
    float* bufA = (float*)d_ws;                     // [NODES*HID]
    float* bufB = bufA + (size_t)NODES * HID;       // [NODES*HID]
    float* dinv = bufB + (size_t)NODES * HID;       // [NODES]
    float* Bp   = dinv + NODES;                     // up to 256*256 floats

    const int B = 256;
    const long long NF = (long long)NODES * HID;
    const long long SC = (long long)EDGES * (HID / 4);

    // degrees (self-loop contributes 1) -> dinv = rsqrt(deg)
    fill_f32<<<(NODES + B - 1) / B, B, 0, stream>>>(dinv, 1.0f, NODES);
    degree_count<<<(EDGES + B - 1) / B, B, 0, stream>>>(dstp, dinv, EDGES);
    rsqrt_inplace<<<(NODES + B - 1) / B, B, 0, stream>>>(dinv, NODES);

    dim3 blk(256);
    dim3 gHID((NODES + 63) / 64, HID / 128);           // 782 x 2
    dim3 gOUT((NODES + 63) / 64, 1);                   // 782 x 1 (Npad=128)

    // ----- layer 1: H1 = x @ W1 -----
    pack_B<<<((FIN / 4) * HID + B - 1) / B, B, 0, stream>>>(W1, Bp, FIN, HID, HID);
    wmma_gemm_f32<<<gHID, blk, FIN * 128 * 4, stream>>>(x, Bp, bufA, NODES, HID, HID, FIN);
    fill_f32<<<(int)((NF + B - 1) / B), B, 0, stream>>>(bufB, 0.0f, NF);
    scatter_edges<<<(int)((SC + B - 1) / B), B, 0, stream>>>(bufA, bufB, srcp, dstp, dinv, EDGES, HID);
    epilogue_relu<<<(int)((NF + B - 1) / B), B, 0, stream>>>(bufB, bufA, dinv, b1, NODES, HID);

    // ----- layer 2: H2 = H1R @ W2 -----
    pack_B<<<((HID / 4) * HID + B - 1) / B, B, 0, stream>>>(W2, Bp, HID, HID, HID);
    wmma_gemm_f32<<<gHID, blk, HID * 128 * 4, stream>>>(bufB, Bp, bufA, NODES, HID, HID, HID);
    fill_f32<<<(int)((NF + B - 1) / B), B, 0, stream>>>(bufB, 0.0f, NF);
    scatter_edges<<<(int)((SC + B - 1) / B), B, 0, stream>>>(bufA, bufB, srcp, dstp, dinv, EDGES, HID);
    epilogue_relu<<<(int)((NF + B - 1) / B), B, 0, stream>>>(bufB, bufA, dinv, b2, NODES, HID);

    // ----- classifier: out = H2R @ Wc + bc  (Npad = 128 covers OUTC=40) -----
    float* out = (float*)d_out;                        // [NODES, OUTC]
    pack_B<<<((HID / 4) * 128 + B - 1) / B, B, 0, stream>>>(Wc, Bp, HID, OUTC, 128);
    wmma_gemm_f32<<<gOUT, blk, HID * 128 * 4, stream>>>(bufB, Bp, out, NODES, OUTC, 128, HID);
    add_bias<<<(NODES * OUTC + B - 1) / B, B, 0, stream>>>(out, bc, NODES * OUTC, OUTC);
}